// PQC_90031104459439
// MI455X (gfx1250) — compile-verified
//
#include <hip/hip_runtime.h>
#include <hip/hip_bf16.h>

// ---------------------------------------------------------------------------
// 14-qubit statevector sim, B=256, 2 circuits.
//  - prep kernel: fuse each rotation layer into four 16x16 complex matrices
//    (Kron of 2x2 gates over wire groups {0-3,4-7,8-11,10-13(I padded)}),
//    stored as [circuit][layer(9)][group(4)][{Re,Im,-Im}][16x16] in d_ws.
//  - sim kernel: one workgroup per (batch, circuit); state (re/im planes)
//    lives in LDS (ping-pong, 256 KB dynamic). Each group application is a
//    16x16 complex GEMM over 1024 columns done with v_wmma_f32_16x16x4_f32
//    (4 real WMMAs per K-chunk: Dr += Ur*Sr + (-Ui)*Si ; Di += Ur*Si + Ui*Sr).
//    Tile loop has a uniform trip count so no exec masking wraps the WMMAs;
//    group loop is unrolled so LDS addressing is compile-time per group.
//    CNOT chain = basis permutation gather. Output = weighted |psi|^2 reduce.
// ---------------------------------------------------------------------------

#define N_Q 14
#define N_AMP (1 << N_Q)          // 16384
#define N_LAYERAPP 9              // 8 layers + final rotation layer
#define N_GROUP 4
#define GATE_STRIDE (3 * 256)     // Re, Im, -Im, each 16x16
#define WS_FLOATS (2 * N_LAYERAPP * N_GROUP * GATE_STRIDE)

typedef float v2f __attribute__((ext_vector_type(2)));
typedef float v8f __attribute__((ext_vector_type(8)));

__device__ __forceinline__ v8f wmma_f32(v2f a, v2f b, v8f c) {
    // (neg_a, A, neg_b, B, c_mod, C, reuse_a, reuse_b)
    return __builtin_amdgcn_wmma_f32_16x16x4_f32(false, a, false, b, (short)0, c,
                                                 false, false);
}

struct cplx { float re, im; };
__device__ __forceinline__ cplx cmul(cplx a, cplx b) {
    return { a.re * b.re - a.im * b.im, a.re * b.im + a.im * b.re };
}

// ------------------------- gate fusion (prep) ------------------------------
__global__ void PQC_prep_gates(const float* __restrict__ qx1,
                               const float* __restrict__ qz1,
                               const float* __restrict__ qx2,
                               const float* __restrict__ qz2,
                               float* __restrict__ gates) {
    const int blk = blockIdx.x;            // 0..71
    const int c   = blk / 36;              // circuit
    const int rem = blk % 36;
    const int L   = rem >> 2;              // layer-app 0..8
    const int g   = rem & 3;               // group 0..3

    const float* qx = (c == 0) ? qx1 : qx2;
    const float* qz = (c == 0) ? qz1 : qz2;

    const int tid = threadIdx.x;           // 256 threads -> one matrix element
    const int mp  = tid >> 4;              // output row m'
    const int m   = tid & 15;              // input  col m

    const int s_of_g[4] = {0, 4, 8, 10};
    const int s = s_of_g[g];

    cplx acc = {1.0f, 0.0f};
    #pragma unroll
    for (int o = 0; o < 4; ++o) {
        const int a = (mp >> (3 - o)) & 1;
        const int b = (m  >> (3 - o)) & 1;
        const bool isgate = (g < 3) || (o >= 2);   // group 3: I on wires 10,11
        cplx e;
        if (isgate) {
            const int w = s + o;
            const float hx = 0.5f * qx[L * N_Q + w];
            const float hz = 0.5f * qz[L * N_Q + w];
            const float cx = cosf(hx), sx = sinf(hx);
            const float cz = cosf(hz), sz = sinf(hz);
            // U = diag(e^{-i hz}, e^{+i hz}) * [[cx, -i sx],[-i sx, cx]]
            if (a == 0 && b == 0)      e = {  cz * cx, -sz * cx };
            else if (a == 0 && b == 1) e = { -sx * sz, -sx * cz };
            else if (a == 1 && b == 0) e = {  sx * sz, -sx * cz };
            else                       e = {  cz * cx,  sz * cx };
        } else {
            e = (a == b) ? cplx{1.0f, 0.0f} : cplx{0.0f, 0.0f};
        }
        acc = cmul(acc, e);
    }

    float* dst = gates + (((c * N_LAYERAPP + L) * N_GROUP + g) * 3) * 256;
    dst[0 * 256 + mp * 16 + m] = acc.re;   // Re(U)
    dst[1 * 256 + mp * 16 + m] = acc.im;   // Im(U)
    dst[2 * 256 + mp * 16 + m] = -acc.im;  // -Im(U)
}

// ----------------------------- simulator -----------------------------------
extern __shared__ float smem[];   // 2 ping-pong buffers: [re 16384][im 16384] x2

__global__ void __launch_bounds__(256)
PQC_simulate(const float* __restrict__ x,
             const float* __restrict__ gates,
             const float* __restrict__ c1,
             const float* __restrict__ c2,
             float* __restrict__ out) {
    const int b    = blockIdx.x;           // batch element
    const int circ = blockIdx.y;           // circuit 0/1
    const int tid  = threadIdx.x;          // 0..255
    const int lane = tid & 31;
    // wave id is uniform per wave: pin it to an SGPR so the tile loop is
    // scalar and no exec masking wraps the WMMA block (EXEC must be all 1s).
    const int wv   = __builtin_amdgcn_readfirstlane(tid >> 5);   // 0..7
    const int mrow  = lane & 15;
    const int khalf = (lane >> 4) << 1;    // 0 or 2 (VGPR-pair K split)

    int cur = 0;                           // element offset of active buffer

    // --- initial basis state from sign bits of x ---
    int k0 = 0;
    #pragma unroll
    for (int w = 0; w < N_Q; ++w)
        k0 |= (x[b * N_Q + w] > 0.0f ? 1 : 0) << (N_Q - 1 - w);

    {
        float* re = smem + cur;
        float* im = smem + cur + N_AMP;
        for (int k = 0; k < N_AMP / 256; ++k) {
            const int i = tid + k * 256;
            re[i] = (i == k0) ? 1.0f : 0.0f;
            im[i] = 0.0f;
        }
    }

    for (int L = 0; L < N_LAYERAPP; ++L) {
        // ---- four grouped 16x16 complex matmuls (all 14 wires rotated) ----
        #pragma unroll
        for (int g = 0; g < N_GROUP; ++g) {
            __syncthreads();
            const int log2R_of_g[4] = {10, 6, 2, 0};
            const int log2R = log2R_of_g[g];           // compile-time per g
            const int Rm1   = (1 << log2R) - 1;
            const float* gw =
                gates + (((circ * N_LAYERAPP + L) * N_GROUP + g) * 3) * 256;

            // A fragments (16x4 per K-chunk): lane m = lane&15, K = kc*4+khalf+j
            v2f aUr[4], aUi[4], aUn[4];
            #pragma unroll
            for (int kc = 0; kc < 4; ++kc) {
                const int k0i = kc * 4 + khalf;
                aUr[kc] = v2f{gw[0 * 256 + mrow * 16 + k0i],
                              gw[0 * 256 + mrow * 16 + k0i + 1]};
                aUi[kc] = v2f{gw[1 * 256 + mrow * 16 + k0i],
                              gw[1 * 256 + mrow * 16 + k0i + 1]};
                aUn[kc] = v2f{gw[2 * 256 + mrow * 16 + k0i],
                              gw[2 * 256 + mrow * 16 + k0i + 1]};
            }

            float* re = smem + cur;
            float* im = smem + cur + N_AMP;

            // 1024 columns -> 64 tiles of 16; 8 tiles per wave, uniform trip
            // count (no divergent exec around the WMMAs). Column-local =>
            // in-place update is safe.
            for (int t = 0; t < 8; ++t) {
                const int tile  = wv + t * 8;
                const int col   = tile * 16 + mrow;
                const int l     = col >> log2R;
                const int r     = col & Rm1;
                const int cbase = (l << (log2R + 4)) + r;

                v2f bSr[4], bSi[4];
                #pragma unroll
                for (int kc = 0; kc < 4; ++kc) {
                    const int m0 = kc * 4 + khalf;
                    const int a0 = cbase + (m0 << log2R);
                    const int a1 = cbase + ((m0 + 1) << log2R);
                    bSr[kc] = v2f{re[a0], re[a1]};
                    bSi[kc] = v2f{im[a0], im[a1]};
                }

                v8f dr = {};
                v8f di = {};
                #pragma unroll
                for (int kc = 0; kc < 4; ++kc) {
                    dr = wmma_f32(aUr[kc], bSr[kc], dr);
                    dr = wmma_f32(aUn[kc], bSi[kc], dr);   // -Ui * Si
                    di = wmma_f32(aUr[kc], bSi[kc], di);
                    di = wmma_f32(aUi[kc], bSr[kc], di);
                }

                #pragma unroll
                for (int v = 0; v < 8; ++v) {
                    const int m = ((lane >> 4) << 3) + v;
                    const int a = cbase + (m << log2R);
                    re[a] = dr[v];
                    im[a] = di[v];
                }
            }
        }

        // ---- CNOT chain (layers 0..7 only): basis permutation gather ----
        if (L < N_LAYERAPP - 1) {
            __syncthreads();
            const float* sre  = smem + cur;
            const float* sim_ = smem + cur + N_AMP;
            const int oth = cur ^ (2 * N_AMP);
            float* dre  = smem + oth;
            float* dim_ = smem + oth + N_AMP;
            for (int k = 0; k < N_AMP / 256; ++k) {
                const int i = tid + k * 256;
                int j = i;                     // walk CNOT(w,w+1), w = 12..0
                #pragma unroll
                for (int cb = 1; cb <= 13; ++cb)
                    j ^= ((j >> cb) & 1) << (cb - 1);
                dre[i]  = sre[j];
                dim_[i] = sim_[j];
            }
            cur = oth;
        }
    }

    // ---- measurement: f = sum_i |psi_i|^2 * S(i), S(i)=sum_w cw[w]*(+/-1) --
    __syncthreads();
    const float* cw = (circ == 0) ? c1 : c2;
    float cwr[N_Q];
    #pragma unroll
    for (int w = 0; w < N_Q; ++w) cwr[w] = cw[w];

    const float* re = smem + cur;
    const float* im = smem + cur + N_AMP;
    float acc = 0.0f;
    for (int k = 0; k < N_AMP / 256; ++k) {
        const int i = tid + k * 256;
        const float pr = re[i] * re[i] + im[i] * im[i];
        float s = 0.0f;
        #pragma unroll
        for (int w = 0; w < N_Q; ++w)
            s += ((i >> (N_Q - 1 - w)) & 1) ? -cwr[w] : cwr[w];
        acc += pr * s;
    }

    float* partial = smem + (cur ^ (2 * N_AMP));   // scratch in inactive buffer
    partial[tid] = acc;
    __syncthreads();
    for (int stride = 128; stride > 0; stride >>= 1) {
        if (tid < stride) partial[tid] += partial[tid + stride];
        __syncthreads();
    }
    if (tid == 0) out[2 * b + circ] = partial[0];
}

// ----------------------------- launcher ------------------------------------
extern "C" void kernel_launch(void* const* d_in, const int* in_sizes, int n_in,
                              void* d_out, int out_size, void* d_ws, size_t ws_size,
                              hipStream_t stream) {
    const float* x   = (const float*)d_in[0];
    const float* qx1 = (const float*)d_in[1];
    const float* qz1 = (const float*)d_in[2];
    const float* c1  = (const float*)d_in[3];
    const float* qx2 = (const float*)d_in[4];
    const float* qz2 = (const float*)d_in[5];
    const float* c2  = (const float*)d_in[6];
    float* out = (float*)d_out;
    float* gates = (float*)d_ws;              // WS_FLOATS floats (~216 KB)

    PQC_prep_gates<<<dim3(2 * N_LAYERAPP * N_GROUP), dim3(256), 0, stream>>>(
        qx1, qz1, qx2, qz2, gates);

    const size_t lds_bytes = 4u * N_AMP * sizeof(float);   // 256 KB ping-pong
    PQC_simulate<<<dim3(256, 2), dim3(256), lds_bytes, stream>>>(
        x, gates, c1, c2, out);
}